// QueryFusion_7859790152234
// MI455X (gfx1250) — compile-verified
//
#include <hip/hip_runtime.h>

typedef __attribute__((ext_vector_type(2))) float v2f;
typedef __attribute__((ext_vector_type(4))) float v4f;
typedef __attribute__((ext_vector_type(8))) float v8f;

#define M_PILLARS 16384
#define N_PTS     4096
#define NTILES    (N_PTS / 16)   // 256 point tiles
#define C_FEAT    64
#define QTILE     128            // queries per block (8 waves x 16)
#define THREADS   256

// Constants from the reference
#define VOX_X     0.16f
#define VOX_Y     0.16f
#define RANGE_X   0.0f
#define RANGE_Y  -25.6f
#define RADIUS2   1.0f           // radius^2

__global__ __launch_bounds__(THREADS)
void ball_query_wmma_fused(const float* __restrict__ pillar,
                           const int*   __restrict__ coors,
                           const float* __restrict__ seg_feats,
                           const float* __restrict__ seg_points,
                           float*       __restrict__ out)
{
    // B fragments staged in EXACT wmma operand layout, one 32-lane x v2f record
    // per 16-point tile:  lane l<16 -> (-2px, -2py),  lane l>=16 -> (1, p2).
    // Inner loop then needs a single branchless ds_load_b64 per tile.
    __shared__ __align__(16) float s_b[NTILES * 64];   // 64 KB
    __shared__ int   s_idx[QTILE][4];
    __shared__ float s_flag[QTILE];

    const int tid = threadIdx.x;

    // Warm caches for the pillar tile consumed in phase C (global_prefetch_b8).
    __builtin_prefetch(&pillar[(size_t)blockIdx.x * QTILE * C_FEAT + tid * 32], 0, 0);

    // ---- Phase A: stage all points into LDS in fragment layout ---------------
    for (int i = tid; i < N_PTS; i += THREADS) {
        float x = seg_points[3 * i + 0];
        float y = seg_points[3 * i + 1];
        int   t   = i >> 4;          // tile
        int   col = i & 15;          // column within tile
        float* dst = &s_b[t * 64];
        dst[col * 2 + 0]        = -2.0f * x;
        dst[col * 2 + 1]        = -2.0f * y;
        dst[(16 + col) * 2 + 0] = 1.0f;
        dst[(16 + col) * 2 + 1] = x * x + y * y;
    }
    __syncthreads();

    // ---- Phase B: per-wave 16-query ball query via WMMA ----------------------
    const int wave = tid >> 5;
    const int lane = tid & 31;
    const int row  = lane & 15;
    const int qrow = blockIdx.x * QTILE + wave * 16 + row;

    // Query geometry: A-row m = (qx, qy, q2, 1)
    const int   cy = coors[4 * qrow + 2];
    const int   cx = coors[4 * qrow + 3];
    const float qx = ((float)cx + 0.5f) * VOX_X + RANGE_X;
    const float qy = ((float)cy + 0.5f) * VOX_Y + RANGE_Y;
    const float q2 = qx * qx + qy * qy;

    // A fragment (16x4 f32): lanes 0-15 hold K=0,1 ; lanes 16-31 hold K=2,3
    v2f a;
    if (lane < 16) { a.x = qx; a.y = qy; }
    else           { a.x = q2; a.y = 1.0f; }

    int cnt = (lane < 16) ? 0 : 4;     // lanes 16-31 start saturated
    int i0 = 0, i1 = 0, i2 = 0, i3 = 0;
    const unsigned r8    = (unsigned)(row & 7);
    const float*   bptr  = &s_b[lane * 2];

    for (int t = 0; t < NTILES; ++t) {
        // Branchless B fragment: one ds_load_b64, conflict-free across 32 lanes.
        v2f b = *(const v2f*)(bptr + t * 64);

        v8f c = {};
        v8f d = __builtin_amdgcn_wmma_f32_16x16x4_f32(
            /*neg_a=*/false, a, /*neg_b=*/false, b,
            /*c_mod=*/(short)0, c, /*reuse_a=*/false, /*reuse_b=*/false);

        // Ballot per output VGPR r: bits0-15 = row r, bits16-31 = row r+8,
        // bit order == point index order.
        unsigned mk[8];
#pragma unroll
        for (int r = 0; r < 8; ++r)
            mk[r] = (unsigned)__ballot(d[r] < RADIUS2);

        unsigned mm = mk[0];
        mm = (r8 == 1u) ? mk[1] : mm;
        mm = (r8 == 2u) ? mk[2] : mm;
        mm = (r8 == 3u) ? mk[3] : mm;
        mm = (r8 == 4u) ? mk[4] : mm;
        mm = (r8 == 5u) ? mk[5] : mm;
        mm = (r8 == 6u) ? mk[6] : mm;
        mm = (r8 == 7u) ? mk[7] : mm;
        unsigned rowmask = (row < 8) ? (mm & 0xFFFFu) : (mm >> 16);
        if (lane >= 16) rowmask = 0u;

        // Pop in-radius columns in scan order (expected ~0-1 bits per tile).
        while (rowmask != 0u && cnt < 4) {
            int j = (t << 4) + __builtin_ctz(rowmask);
            if      (cnt == 0) i0 = j;
            else if (cnt == 1) i1 = j;
            else if (cnt == 2) i2 = j;
            else               i3 = j;
            ++cnt;
            rowmask &= rowmask - 1u;
        }

        // Wave-uniform early exit once all 16 rows found 4 neighbors.
        unsigned done = (unsigned)__ballot(cnt >= 4);
        if (__builtin_amdgcn_readfirstlane((int)done) == -1) break;
    }

    // Finalize per reference: pad slots with first index; all-zero if none.
    if (lane < 16) {
        int first = (cnt >= 1) ? i0 : 0;
        i0 = first;
        i1 = (cnt >= 2) ? i1 : first;
        i2 = (cnt >= 3) ? i2 : first;
        i3 = (cnt >= 4) ? i3 : first;
        float fl = ((i0 + i1 + i2 + i3) > 0) ? 1.0f : 0.0f;
        int lr = wave * 16 + row;
        s_idx[lr][0] = i0;
        s_idx[lr][1] = i1;
        s_idx[lr][2] = i2;
        s_idx[lr][3] = i3;
        s_flag[lr]   = fl;
    }
    __syncthreads();

    // ---- Phase C: gather 4 neighbor features, max-pool, add (float4) --------
    const int base = blockIdx.x * QTILE;
    for (int e = tid * 4; e < QTILE * C_FEAT; e += THREADS * 4) {
        int lr = e >> 6;      // local row
        int ch = e & 63;      // channel (multiple of 4 -> 16B aligned)
        int j0 = s_idx[lr][0], j1 = s_idx[lr][1];
        int j2 = s_idx[lr][2], j3 = s_idx[lr][3];
        float fl = s_flag[lr];
        v4f p0 = *(const v4f*)&seg_feats[j0 * C_FEAT + ch];
        v4f p1 = *(const v4f*)&seg_feats[j1 * C_FEAT + ch];
        v4f p2 = *(const v4f*)&seg_feats[j2 * C_FEAT + ch];
        v4f p3 = *(const v4f*)&seg_feats[j3 * C_FEAT + ch];
        size_t g = (size_t)(base + lr) * C_FEAT + ch;
        v4f pf = *(const v4f*)&pillar[g];
        v4f r;
#pragma unroll
        for (int k = 0; k < 4; ++k) {
            float pooled = fmaxf(fmaxf(p0[k], p1[k]), fmaxf(p2[k], p3[k]));
            r[k] = pf[k] + pooled * fl;
        }
        *(v4f*)&out[g] = r;
    }
}

extern "C" void kernel_launch(void* const* d_in, const int* in_sizes, int n_in,
                              void* d_out, int out_size, void* d_ws, size_t ws_size,
                              hipStream_t stream) {
    const float* pillar     = (const float*)d_in[0];   // (M, 64) f32
    const int*   coors      = (const int*)  d_in[1];   // (M, 4)  i32
    const float* seg_feats  = (const float*)d_in[2];   // (N, 64) f32
    const float* seg_points = (const float*)d_in[3];   // (N, 3)  f32
    float* out = (float*)d_out;                        // (M, 64) f32

    dim3 grid(M_PILLARS / QTILE);   // 128 blocks
    dim3 block(THREADS);            // 8 waves of 32
    hipLaunchKernelGGL(ball_query_wmma_fused, grid, block, 0, stream,
                       pillar, coors, seg_feats, seg_points, out);
}